// HybridQgnnModel_63866163692279
// MI455X (gfx1250) — compile-verified
//
#include <hip/hip_runtime.h>
#include <hip/hip_bf16.h>

// ---------------------------------------------------------------------------
// Hybrid quantum-GNN on MI455X (gfx1250, wave32).
// State vector: 2^19 complex amps stored SoA (re[],im[]) in workspace.
// Per circuit layer: two LDS-resident phases; fused 4-qubit 16x16 complex
// unitaries applied with V_WMMA_F32_16X16X4_F32 chains; CNOT chains applied
// as an index permutation during writeback.
// ---------------------------------------------------------------------------

typedef float v2f __attribute__((ext_vector_type(2)));
typedef float v8f __attribute__((ext_vector_type(8)));

#if defined(__HIP_DEVICE_COMPILE__) && defined(__gfx1250__) && \
    __has_builtin(__builtin_amdgcn_wmma_f32_16x16x4_f32)
#define USE_WMMA_F32 1
#else
#define USE_WMMA_F32 0
#endif

#define NQ 19
#define STATE_N (1 << NQ)          // 524288 amplitudes
#define CHUNK_BITS 13
#define CHUNK (1 << CHUNK_BITS)    // 8192 amps per workgroup (64KB LDS SoA)
#define N_LAYERS 7                 // 1 feature map + 6 ansatz layers
#define MATS_PER_LAYER 6           // 4 phase-A groups + 2 phase-B groups
#define MAT_FLOATS 512             // 16x16 re + 16x16 im

// workspace layout (floats)
#define WS_FEATS   0
#define WS_MATS    32
#define WS_SRE     (WS_MATS + N_LAYERS * MATS_PER_LAYER * MAT_FLOATS)  // 32+21504
#define WS_SIM     (WS_SRE + STATE_N)
#define WS_PART    (WS_SIM + STATE_N)

// ---------------------------------------------------------------------------
// 1) state init
// ---------------------------------------------------------------------------
__global__ void init_state_kernel(float* __restrict__ re, float* __restrict__ im) {
    int i = blockIdx.x * blockDim.x + threadIdx.x;
    re[i] = (i == 0) ? 1.0f : 0.0f;
    im[i] = 0.0f;
}

// ---------------------------------------------------------------------------
// 2) GCN front-end -> feats[19]  (tiny: 8 nodes / 8 edges, one thread)
// ---------------------------------------------------------------------------
__global__ void gcn_feats_kernel(const int* __restrict__ eidx,
                                 const float* __restrict__ state,
                                 const float* __restrict__ mass,
                                 const float* __restrict__ spin,
                                 const float* __restrict__ charge,
                                 const float* __restrict__ p_norm,
                                 const float* __restrict__ theta,
                                 const float* __restrict__ gnw, const float* __restrict__ gnb,
                                 const float* __restrict__ gew, const float* __restrict__ geb,
                                 const float* __restrict__ node_w, const float* __restrict__ node_b,
                                 const float* __restrict__ edge_w, const float* __restrict__ edge_b,
                                 float* __restrict__ feats) {
    if (threadIdx.x != 0 || blockIdx.x != 0) return;
    int src[16], dst[16];
    for (int e = 0; e < 8; ++e) { src[e] = eidx[e]; dst[e] = eidx[8 + e]; }
    for (int n = 0; n < 8; ++n) { src[8 + n] = n; dst[8 + n] = n; }  // self loops
    float deg[8];
    for (int n = 0; n < 8; ++n) deg[n] = 0.f;
    for (int e = 0; e < 16; ++e) deg[dst[e]] += 1.f;
    float dis[8];
    for (int n = 0; n < 8; ++n) dis[n] = (deg[n] > 0.f) ? rsqrtf(deg[n]) : 0.f;

    // node conv: (8,4)@(4,1)
    float h[8];
    for (int n = 0; n < 8; ++n) {
        float a = 0.f;
        for (int f = 0; f < 4; ++f) a += state[n * 4 + f] * gnw[f];
        h[n] = a;
    }
    float outn[8];
    for (int n = 0; n < 8; ++n) outn[n] = 0.f;
    for (int e = 0; e < 16; ++e) outn[dst[e]] += h[src[e]] * dis[src[e]] * dis[dst[e]];
    for (int n = 0; n < 8; ++n) outn[n] += gnb[0];
    // edge conv: edge feats (mass,spin,charge) on same 8-node index space
    float he[8];
    for (int n = 0; n < 8; ++n)
        he[n] = mass[n] * gew[0] + spin[n] * gew[1] + charge[n] * gew[2];
    float oute[8];
    for (int n = 0; n < 8; ++n) oute[n] = 0.f;
    for (int e = 0; e < 16; ++e) oute[dst[e]] += he[src[e]] * dis[src[e]] * dis[dst[e]];
    for (int n = 0; n < 8; ++n) oute[n] += geb[0];

    for (int k = 0; k < 9; ++k) {
        float a = node_b[k];
        for (int n = 0; n < 8; ++n) a += outn[n] * node_w[n * 9 + k];
        feats[k] = a;
    }
    for (int k = 0; k < 8; ++k) {
        float a = edge_b[k];
        for (int n = 0; n < 8; ++n) a += oute[n] * edge_w[n * 8 + k];
        feats[9 + k] = a;
    }
    feats[17] = p_norm[0];
    feats[18] = theta[0];
}

// ---------------------------------------------------------------------------
// 3) build fused 16x16 complex unitaries (Kronecker of four 2x2 gates)
//    layer 0: G = RZ(feats[q]) * H ; layers 1..6: G = RZ(w) * RX(w)
//    stored transposed (B[K=j][N=i] = U[i][j]) for direct WMMA B loads.
// ---------------------------------------------------------------------------
__device__ __forceinline__ void gate2x2(int layer, int q,
                                        const float* feats, const float* qw,
                                        float ur[2][2], float ui[2][2]) {
    if (layer == 0) {
        float f = feats[q];
        float c = __cosf(0.5f * f), s = __sinf(0.5f * f);
        const float inv = 0.70710678118654752f;
        ur[0][0] = c * inv;  ui[0][0] = -s * inv;
        ur[0][1] = c * inv;  ui[0][1] = -s * inv;
        ur[1][0] = c * inv;  ui[1][0] = s * inv;
        ur[1][1] = -c * inv; ui[1][1] = -s * inv;
    } else {
        int blk = (layer - 1) / 3, j = (layer - 1) % 3;
        float t = qw[blk * (NQ * 3) + q * 3 + j];
        float c = __cosf(0.5f * t), s = __sinf(0.5f * t);
        // G = RZ(t) * RX(t)
        ur[0][0] = c * c;   ui[0][0] = -s * c;
        ur[0][1] = -s * s;  ui[0][1] = -c * s;
        ur[1][0] = s * s;   ui[1][0] = -c * s;
        ur[1][1] = c * c;   ui[1][1] = s * c;
    }
}

__global__ void build_mats_kernel(const float* __restrict__ feats,
                                  const float* __restrict__ qw,
                                  float* __restrict__ mats) {
    // factor qubit per (group, slot); -1 => identity
    // phase A groups 0..3 at local bits p=0,4,8,9; phase B groups 4,5 at p=6,8
    const int fq[6][4] = {
        {0, 1, 2, 3}, {4, 5, 6, 7}, {8, 9, 10, 11}, {-1, -1, -1, 12},
        {13, 14, 15, 16}, {-1, -1, 17, 18}};
    int matId = blockIdx.x;            // 0..41
    int layer = matId / MATS_PER_LAYER;
    int g = matId % MATS_PER_LAYER;
    int t = threadIdx.x;               // 0..255
    int i = t & 15, j = t >> 4;
    float pr = 1.f, pi = 0.f;
    for (int s = 0; s < 4; ++s) {
        int q = fq[g][s];
        float ur[2][2], ui[2][2];
        if (q < 0) {
            ur[0][0] = 1.f; ur[0][1] = 0.f; ur[1][0] = 0.f; ur[1][1] = 1.f;
            ui[0][0] = 0.f; ui[0][1] = 0.f; ui[1][0] = 0.f; ui[1][1] = 0.f;
        } else {
            gate2x2(layer, q, feats, qw, ur, ui);
        }
        int ib = (i >> s) & 1, jb = (j >> s) & 1;
        float gr = ur[ib][jb], gi = ui[ib][jb];
        float nr = pr * gr - pi * gi;
        float ni = pr * gi + pi * gr;
        pr = nr; pi = ni;
    }
    mats[matId * MAT_FLOATS + j * 16 + i] = pr;        // re, transposed (j row)
    mats[matId * MAT_FLOATS + 256 + j * 16 + i] = pi;  // im
}

// ---------------------------------------------------------------------------
// 4) gate kernel: one 2^13-amp chunk per workgroup in LDS; WMMA group passes;
//    CNOT-chain permutation folded into writeback.
// ---------------------------------------------------------------------------
__device__ __forceinline__ void apply_group(float* lre, float* lim,
                                            const float* __restrict__ gm, int p) {
    const int lane = threadIdx.x & 31;
    const int wave = threadIdx.x >> 5;
    const int lh = lane >> 4;     // lane half (0/1)
    const int lr = lane & 15;     // row-in-tile (A) / N column (B,C)
    const int pm = (1 << p) - 1;
#if USE_WMMA_F32
    // B = U^T: lane-resident, reused across all tiles.  VGPR v of k-step kc
    // holds B[K = kc*4 + v + 2*lh][N = lr]  (mirrors documented A layout).
    v2f br[4], bi[4];
#pragma unroll
    for (int kc = 0; kc < 4; ++kc) {
#pragma unroll
        for (int v = 0; v < 2; ++v) {
            int K = kc * 4 + v + 2 * lh;
            int idx = K * 16 + lr;
            br[kc][v] = gm[idx];
            bi[kc][v] = gm[256 + idx];
        }
    }
    // 512 composite rows per chunk -> 8 waves * 4 tiles of 16 rows
    for (int tIdx = 0; tIdx < 4; ++tIdx) {
        int m0 = wave * 64 + tIdx * 16;
        v8f P = {}, Q = {}, R = {}, S = {};
#pragma unroll
        for (int kc = 0; kc < 4; ++kc) {
            int mA = m0 + lr;
            int base = ((mA >> p) << (p + 4)) + (mA & pm);
            v2f ar, ai;
#pragma unroll
            for (int v = 0; v < 2; ++v) {
                int j = kc * 4 + v + 2 * lh;
                int l = base + (j << p);
                ar[v] = lre[l];
                ai[v] = lim[l];
            }
            // complex matmul via 4 real WMMAs: Dre = P - Q, Dim = R + S
            P = __builtin_amdgcn_wmma_f32_16x16x4_f32(false, ar, false, br[kc], (short)0, P, false, false);
            Q = __builtin_amdgcn_wmma_f32_16x16x4_f32(false, ai, false, bi[kc], (short)0, Q, false, false);
            R = __builtin_amdgcn_wmma_f32_16x16x4_f32(false, ar, false, bi[kc], (short)0, R, false, false);
            S = __builtin_amdgcn_wmma_f32_16x16x4_f32(false, ai, false, br[kc], (short)0, S, false, false);
        }
#pragma unroll
        for (int r = 0; r < 8; ++r) {
            int mD = m0 + r + 8 * lh;
            int base = ((mD >> p) << (p + 4)) + (mD & pm);
            int l = base + (lr << p);
            lre[l] = P[r] - Q[r];
            lim[l] = R[r] + S[r];
        }
    }
#else
    // scalar fallback: each thread owns 2 composite rows (read row, write row)
    for (int rr = 0; rr < 2; ++rr) {
        int m = threadIdx.x * 2 + rr;
        int base = ((m >> p) << (p + 4)) + (m & pm);
        float xr[16], xi[16];
        for (int j = 0; j < 16; ++j) {
            int l = base + (j << p);
            xr[j] = lre[l];
            xi[j] = lim[l];
        }
        for (int i = 0; i < 16; ++i) {
            float accR = 0.f, accI = 0.f;
            for (int j = 0; j < 16; ++j) {
                float urv = gm[j * 16 + i], uiv = gm[256 + j * 16 + i];
                accR += xr[j] * urv - xi[j] * uiv;
                accI += xr[j] * uiv + xi[j] * urv;
            }
            int l = base + (i << p);
            lre[l] = accR;
            lim[l] = accI;
        }
    }
#endif
}

__global__ __launch_bounds__(256) void gate_phase_kernel(float* __restrict__ sre,
                                                         float* __restrict__ sim,
                                                         const float* __restrict__ matsLayer,
                                                         int phase) {
    __shared__ float lre[CHUNK];
    __shared__ float lim[CHUNK];
    const int tid = threadIdx.x;
    const int c = blockIdx.x;   // 64 chunks

    // --- load chunk ---
    for (int k = 0; k < CHUNK / 256; ++k) {
        int l = tid + k * 256;
        int g = (phase == 0) ? ((c << CHUNK_BITS) + l)
                             : ((l & 1) | (c << 1) | ((l >> 1) << 7));
        lre[l] = sre[g];
        lim[l] = sim[g];
    }
    __syncthreads();

    // --- fused 4-qubit group passes (WMMA) ---
    if (phase == 0) {
        const int pList[4] = {0, 4, 8, 9};
        for (int gg = 0; gg < 4; ++gg) {
            apply_group(lre, lim, matsLayer + gg * MAT_FLOATS, pList[gg]);
            __syncthreads();
        }
    } else {
        const int pList[2] = {6, 8};
        for (int gg = 0; gg < 2; ++gg) {
            apply_group(lre, lim, matsLayer + (4 + gg) * MAT_FLOATS, pList[gg]);
            __syncthreads();
        }
    }

    // --- CNOT-chain permutation folded into writeback: out[g(l)] = lds[sigma^-1(l)] ---
    for (int k = 0; k < CHUNK / 256; ++k) {
        int l = tid + k * 256;
        int srcIdx, g;
        if (phase == 0) {
            srcIdx = (l ^ (l << 1)) & 0x1FFF;                       // CNOTs 0->1 .. 11->12
            g = (c << CHUNK_BITS) + l;
        } else {
            srcIdx = l ^ ((l << 1) & 0x0FC0) ^ ((l >> 11) & 1);     // CNOTs 12->13 .. 18->0
            g = (l & 1) | (c << 1) | ((l >> 1) << 7);
        }
        sre[g] = lre[srcIdx];
        sim[g] = lim[srcIdx];
    }
}

// ---------------------------------------------------------------------------
// 5) <Z_0> reduction (deterministic two-stage)
// ---------------------------------------------------------------------------
__global__ void reduce1_kernel(const float* __restrict__ sre,
                               const float* __restrict__ sim,
                               float* __restrict__ partials) {
    __shared__ float sb[256];
    int t = threadIdx.x;
    int gid = blockIdx.x * 256 + t;
    float acc = 0.f;
    for (int i = gid; i < STATE_N; i += 64 * 256) {
        float r = sre[i], m = sim[i];
        float p = r * r + m * m;
        acc += (i & 1) ? -p : p;    // qubit 0 = bit 0 of amplitude index
    }
    sb[t] = acc;
    __syncthreads();
    for (int s = 128; s > 0; s >>= 1) {
        if (t < s) sb[t] += sb[t + s];
        __syncthreads();
    }
    if (t == 0) partials[blockIdx.x] = sb[0];
}

__global__ void reduce2_kernel(const float* __restrict__ partials,
                               float* __restrict__ out) {
    __shared__ float sb[64];
    int t = threadIdx.x;
    sb[t] = partials[t];
    __syncthreads();
    for (int s = 32; s > 0; s >>= 1) {
        if (t < s) sb[t] += sb[t + s];
        __syncthreads();
    }
    if (t == 0) out[0] = sb[0];
}

// ---------------------------------------------------------------------------
// launch
// ---------------------------------------------------------------------------
extern "C" void kernel_launch(void* const* d_in, const int* in_sizes, int n_in,
                              void* d_out, int out_size, void* d_ws, size_t ws_size,
                              hipStream_t stream) {
    const int*   edge_index = (const int*)  d_in[0];
    const float* state      = (const float*)d_in[1];
    const float* mass       = (const float*)d_in[2];
    const float* spin       = (const float*)d_in[3];
    const float* charge     = (const float*)d_in[4];
    const float* p_norm     = (const float*)d_in[5];
    const float* theta      = (const float*)d_in[6];
    // d_in[7] = scattering (unused by reference)
    const float* gcn_node_w = (const float*)d_in[8];
    const float* gcn_node_b = (const float*)d_in[9];
    const float* gcn_edge_w = (const float*)d_in[10];
    const float* gcn_edge_b = (const float*)d_in[11];
    const float* node_w     = (const float*)d_in[12];
    const float* node_b     = (const float*)d_in[13];
    const float* edge_w     = (const float*)d_in[14];
    const float* edge_b     = (const float*)d_in[15];
    const float* qweights   = (const float*)d_in[16];

    float* ws       = (float*)d_ws;
    float* feats    = ws + WS_FEATS;
    float* mats     = ws + WS_MATS;
    float* sre      = ws + WS_SRE;
    float* sim      = ws + WS_SIM;
    float* partials = ws + WS_PART;

    init_state_kernel<<<STATE_N / 256, 256, 0, stream>>>(sre, sim);
    gcn_feats_kernel<<<1, 32, 0, stream>>>(edge_index, state, mass, spin, charge,
                                           p_norm, theta,
                                           gcn_node_w, gcn_node_b, gcn_edge_w, gcn_edge_b,
                                           node_w, node_b, edge_w, edge_b, feats);
    build_mats_kernel<<<N_LAYERS * MATS_PER_LAYER, 256, 0, stream>>>(feats, qweights, mats);

    for (int L = 0; L < N_LAYERS; ++L) {
        const float* ml = mats + L * MATS_PER_LAYER * MAT_FLOATS;
        gate_phase_kernel<<<STATE_N / CHUNK, 256, 0, stream>>>(sre, sim, ml, 0);
        gate_phase_kernel<<<STATE_N / CHUNK, 256, 0, stream>>>(sre, sim, ml, 1);
    }

    reduce1_kernel<<<64, 256, 0, stream>>>(sre, sim, partials);
    reduce2_kernel<<<1, 64, 0, stream>>>(partials, (float*)d_out);
}